// FCGF_AVG2_89575837925684
// MI455X (gfx1250) — compile-verified
//
#include <hip/hip_runtime.h>
#include <hip/hip_bf16.h>

typedef __attribute__((ext_vector_type(2))) float v2f;
typedef __attribute__((ext_vector_type(8))) float v8f;

#define BN_EPS 1e-5f

// ---------------------------------------------------------------------------
// Kernel 0: exclusive prefix sum of segment lengths (B = 512)
// ---------------------------------------------------------------------------
__global__ __launch_bounds__(512) void seg_offsets_kernel(const int* __restrict__ len,
                                                          int* __restrict__ offs) {
    __shared__ int l[512];
    const int t = threadIdx.x;
    l[t] = len[t];
    __syncthreads();
    int s = 0;
    for (int i = 0; i < t; ++i) s += l[i];
    offs[t] = s;
}

__device__ inline float4 f4add(float4 a, float4 b) {
    return make_float4(a.x + b.x, a.y + b.y, a.z + b.z, a.w + b.w);
}

// ---------------------------------------------------------------------------
// Kernel 1: per-segment mean of x[N,32]  (bandwidth-bound: 512 MB streamed)
// One block per segment; float4 loads -> each wave reads 512 contiguous bytes.
// ---------------------------------------------------------------------------
__global__ __launch_bounds__(256) void seg_mean_kernel(const float* __restrict__ x,
                                                       const int* __restrict__ len,
                                                       const int* __restrict__ offs,
                                                       float* __restrict__ mean) {
    const int b = blockIdx.x;
    const int t = threadIdx.x;
    const int col4 = t & 7;   // which float4 of the 32-float row
    const int rg   = t >> 3;  // row group 0..31
    const long start = offs[b];
    const int  L     = len[b];
    const float4* __restrict__ x4 = (const float4*)x;

    float4 acc = make_float4(0.f, 0.f, 0.f, 0.f);
    for (long r = start + rg; r < start + (long)L; r += 32) {
        float4 v = x4[r * 8 + col4];
        acc = f4add(acc, v);
    }

    __shared__ float4 s[256];
    s[t] = acc;                                   __syncthreads();
    if (t < 128) s[t] = f4add(s[t], s[t + 128]);  __syncthreads();
    if (t <  64) s[t] = f4add(s[t], s[t +  64]);  __syncthreads();
    if (t <  32) s[t] = f4add(s[t], s[t +  32]);  __syncthreads();
    if (t <  16) s[t] = f4add(s[t], s[t +  16]);  __syncthreads();
    if (t < 8) {
        float4 v = f4add(s[t], s[t + 8]);
        const float inv = 1.0f / (float)L;
        mean[b * 32 + t * 4 + 0] = v.x * inv;
        mean[b * 32 + t * 4 + 1] = v.y * inv;
        mean[b * 32 + t * 4 + 2] = v.z * inv;
        mean[b * 32 + t * 4 + 3] = v.w * inv;
    }
}

// ---------------------------------------------------------------------------
// Kernel 2: fused  ReLU(BN(mean@W1+b1))@W2+b2 -> BN -> out
// Single workgroup (512 threads = 16 wave32) so batch-dim BN reductions are
// plain workgroup barriers. GEMMs use V_WMMA_F32_16X16X4_F32 (fp32-exact).
// h[512,64] lives in LDS (stride 66 floats to avoid bank conflicts on the
// A-fragment column loads); t2[512,128] spills to global scratch.
// ---------------------------------------------------------------------------
#define H_STRIDE 66

__global__ __launch_bounds__(512) void mlp_bn_kernel(const float* __restrict__ mean,
                                                     const float* __restrict__ W1,
                                                     const float* __restrict__ b1,
                                                     const float* __restrict__ g1,
                                                     const float* __restrict__ be1,
                                                     const float* __restrict__ W2,
                                                     const float* __restrict__ b2,
                                                     const float* __restrict__ g2,
                                                     const float* __restrict__ be2,
                                                     float* __restrict__ t2,
                                                     float* __restrict__ out) {
    __shared__ float h[512 * H_STRIDE];   // 135168 B
    __shared__ float redS[512];
    __shared__ float redQ[512];
    __shared__ float sc[128];
    __shared__ float sh[128];

    const int t      = threadIdx.x;
    const int wave   = t >> 5;
    const int lane   = t & 31;
    const int l16    = lane & 15;
    const int hiHalf = lane >> 4;   // 0: K0/K1 half, 1: K2/K3 half

    // ---- GEMM1: mean[512,32] @ W1[32,64] + b1 -> h (LDS), t1 kept with bias
    for (int i = 0; i < 8; ++i) {
        const int tile = wave * 8 + i;           // 128 tiles (32 m x 4 n)
        const int m0 = (tile >> 2) * 16;
        const int n0 = (tile & 3) * 16;
        v8f acc = {};
        for (int k = 0; k < 32; k += 4) {
            const int ka = k + hiHalf * 2;
            v2f a, bb;
            a.x  = mean[(m0 + l16) * 32 + ka];
            a.y  = mean[(m0 + l16) * 32 + ka + 1];
            bb.x = W1[ka * 64 + n0 + l16];
            bb.y = W1[(ka + 1) * 64 + n0 + l16];
            acc = __builtin_amdgcn_wmma_f32_16x16x4_f32(false, a, false, bb,
                                                        (short)0, acc, false, false);
        }
        const float bias = b1[n0 + l16];
#pragma unroll
        for (int v = 0; v < 8; ++v) {
            const int row = m0 + v + hiHalf * 8;
            h[row * H_STRIDE + n0 + l16] = acc[v] + bias;
        }
    }
    __syncthreads();

    // ---- BN1 (training stats over 512 rows, 64 cols) + ReLU, in place
    {
        const int col = t & 63, grp = t >> 6;    // 8 row-groups of 64
        float s = 0.f, q = 0.f;
        for (int r = grp * 64; r < grp * 64 + 64; ++r) {
            const float v = h[r * H_STRIDE + col];
            s += v; q += v * v;
        }
        redS[t] = s; redQ[t] = q;                              __syncthreads();
        if (t < 256) { redS[t] += redS[t + 256]; redQ[t] += redQ[t + 256]; } __syncthreads();
        if (t < 128) { redS[t] += redS[t + 128]; redQ[t] += redQ[t + 128]; } __syncthreads();
        if (t < 64) {
            const float s2 = redS[t] + redS[t + 64];
            const float q2 = redQ[t] + redQ[t + 64];
            const float mu  = s2 * (1.0f / 512.0f);
            const float var = q2 * (1.0f / 512.0f) - mu * mu;
            const float scale = g1[t] * rsqrtf(var + BN_EPS);
            sc[t] = scale;
            sh[t] = be1[t] - mu * scale;
        }
        __syncthreads();
        for (int r = grp * 64; r < grp * 64 + 64; ++r) {
            const float v = h[r * H_STRIDE + col] * sc[col] + sh[col];
            h[r * H_STRIDE + col] = v > 0.f ? v : 0.f;
        }
    }
    __syncthreads();

    // ---- GEMM2: h[512,64] @ W2[64,128] + b2 -> t2 (global scratch)
    for (int i = 0; i < 16; ++i) {
        const int tile = wave * 16 + i;          // 256 tiles (32 m x 8 n)
        const int m0 = (tile >> 3) * 16;
        const int n0 = (tile & 7) * 16;
        v8f acc = {};
        for (int k = 0; k < 64; k += 4) {
            const int ka = k + hiHalf * 2;
            v2f a, bb;
            a.x  = h[(m0 + l16) * H_STRIDE + ka];
            a.y  = h[(m0 + l16) * H_STRIDE + ka + 1];
            bb.x = W2[ka * 128 + n0 + l16];
            bb.y = W2[(ka + 1) * 128 + n0 + l16];
            acc = __builtin_amdgcn_wmma_f32_16x16x4_f32(false, a, false, bb,
                                                        (short)0, acc, false, false);
        }
        const float bias = b2[n0 + l16];
#pragma unroll
        for (int v = 0; v < 8; ++v) {
            const int row = m0 + v + hiHalf * 8;
            t2[row * 128 + n0 + l16] = acc[v] + bias;
        }
    }
    __threadfence();   // make our own global stores visible before re-reading
    __syncthreads();

    // ---- BN2 over t2[512,128] -> out
    {
        const int col = t & 127, grp = t >> 7;   // 4 row-groups of 128
        float s = 0.f, q = 0.f;
        for (int r = grp * 128; r < grp * 128 + 128; ++r) {
            const float v = t2[r * 128 + col];
            s += v; q += v * v;
        }
        redS[t] = s; redQ[t] = q;                              __syncthreads();
        if (t < 256) { redS[t] += redS[t + 256]; redQ[t] += redQ[t + 256]; } __syncthreads();
        if (t < 128) {
            const float s2 = redS[t] + redS[t + 128];
            const float q2 = redQ[t] + redQ[t + 128];
            const float mu  = s2 * (1.0f / 512.0f);
            const float var = q2 * (1.0f / 512.0f) - mu * mu;
            const float scale = g2[t] * rsqrtf(var + BN_EPS);
            sc[t] = scale;
            sh[t] = be2[t] - mu * scale;
        }
        __syncthreads();
        for (int r = grp * 128; r < grp * 128 + 128; ++r) {
            out[r * 128 + col] = t2[r * 128 + col] * sc[col] + sh[col];
        }
    }
}

// ---------------------------------------------------------------------------
extern "C" void kernel_launch(void* const* d_in, const int* in_sizes, int n_in,
                              void* d_out, int out_size, void* d_ws, size_t ws_size,
                              hipStream_t stream) {
    const float* x    = (const float*)d_in[0];
    const int*   len  = (const int*)d_in[1];
    const float* W1   = (const float*)d_in[2];
    const float* b1   = (const float*)d_in[3];
    const float* g1   = (const float*)d_in[4];
    const float* be1  = (const float*)d_in[5];
    const float* W2   = (const float*)d_in[6];
    const float* b2   = (const float*)d_in[7];
    const float* g2   = (const float*)d_in[8];
    const float* be2  = (const float*)d_in[9];

    char* ws = (char*)d_ws;
    int*   offs = (int*)ws;                         //   4 KB (512 ints)
    float* mean = (float*)(ws + 4096);              //  64 KB (512*32 f32)
    float* t2   = (float*)(ws + 4096 + 65536);      // 256 KB (512*128 f32)

    seg_offsets_kernel<<<1, 512, 0, stream>>>(len, offs);
    seg_mean_kernel<<<512, 256, 0, stream>>>(x, len, offs, mean);
    mlp_bn_kernel<<<1, 512, 0, stream>>>(mean, W1, b1, g1, be1, W2, b2, g2, be2,
                                         t2, (float*)d_out);
}